// BaseModel_80358838108241
// MI455X (gfx1250) — compile-verified
//
#include <hip/hip_runtime.h>
#include <math.h>

typedef __attribute__((ext_vector_type(2))) float v2f;
typedef __attribute__((ext_vector_type(8))) float v8f;

#define BS     64
#define FD     256
#define MD     128
#define NBIN   8
#define KC     16            // k-rows staged per chunk
#define STRIDE 136           // padded LDS row stride (words): 2*136 % 64 == 16 -> conflict-free
#define CHW    (KC * STRIDE) // words per staged matrix chunk
#define NCHUNK (FD / KC)

// CDNA5 async global->LDS copy (bypasses VGPRs, tracked by ASYNCcnt).
__device__ __forceinline__ void async_ld16(unsigned ldsAddr, const float* g) {
    asm volatile("global_load_async_to_lds_b128 %0, %1, off"
                 :: "v"(ldsAddr), "v"((unsigned long long)g)
                 : "memory");
}

__device__ __forceinline__ void wait_async0() {
    asm volatile("s_wait_asynccnt 0x0" ::: "memory");
}

// Load 4 A-fragments + 4 B-fragments for one k-step: each ds_load_2addr_b32
// lands a fragment's (.x,.y) pair (136 words apart) directly in a VGPR pair,
// so no v_mov shuffling is needed before the WMMAs. One dscnt wait for all 8.
__device__ __forceinline__ void frag_ld(unsigned addrA, unsigned addrB,
                                        v2f& a0, v2f& a1, v2f& a2, v2f& a3,
                                        v2f& b0, v2f& b1, v2f& b2, v2f& b3) {
    asm volatile(
        "ds_load_2addr_b32 %0, %8 offset0:0  offset1:136\n\t"
        "ds_load_2addr_b32 %1, %8 offset0:16 offset1:152\n\t"
        "ds_load_2addr_b32 %2, %8 offset0:32 offset1:168\n\t"
        "ds_load_2addr_b32 %3, %8 offset0:48 offset1:184\n\t"
        "ds_load_2addr_b32 %4, %9 offset0:0  offset1:136\n\t"
        "ds_load_2addr_b32 %5, %9 offset0:16 offset1:152\n\t"
        "ds_load_2addr_b32 %6, %9 offset0:32 offset1:168\n\t"
        "ds_load_2addr_b32 %7, %9 offset0:48 offset1:184\n\t"
        "s_wait_dscnt 0x0"
        : "=&v"(a0), "=&v"(a1), "=&v"(a2), "=&v"(a3),
          "=&v"(b0), "=&v"(b1), "=&v"(b2), "=&v"(b3)
        : "v"(addrA), "v"(addrB));
}

// One workgroup (4 wave32 waves) per (input t, pair a*64+b).
// Waves own 64x64 quadrants: 4x4 WMMA f32 tiles (128 acc VGPRs/lane).
// A/B staged through LDS in double-buffered K-chunks via async-to-LDS loads,
// so each operand element leaves L2 exactly once per workgroup.
// Full 128x128 score tile stays in registers; histogram fully fused.
__global__ __launch_bounds__(128) void gram_hist_kernel(
    const float* __restrict__ in0, const float* __restrict__ in1,
    float* __restrict__ out)
{
    __shared__ float lds[4 * CHW]; // [buf0:A,B][buf1:A,B]
    __shared__ float sMin[4];
    __shared__ float sMax[4];
    __shared__ int   sBins[NBIN];

    const int tid    = threadIdx.x;
    const int lane   = tid & 31;
    const int w      = tid >> 5;   // wave 0..3
    const int laneLo = lane & 15;
    const int laneHi = lane >> 4;
    const int wm     = w & 1;      // M half (rows 64*wm .. +64)
    const int wn     = w >> 1;     // N half (cols 64*wn .. +64)

    const int bid  = blockIdx.x;
    const int t    = bid >> 12;
    const int p    = bid & 4095;
    const int aIdx = p >> 6;
    const int bIdx = p & 63;

    const float* __restrict__ src = t ? in1 : in0;
    const float* __restrict__ A   = src + (size_t)aIdx * FD * MD;
    const float* __restrict__ B   = src + (size_t)bIdx * FD * MD;

    // Staging map: thread covers 16 consecutive floats of a chunk row (4 x b128).
    const int srow = tid >> 3;         // 0..15
    const int scol = (tid & 7) * 16;   // 0..112
    const unsigned stW = (unsigned)(srow * STRIDE + scol);

    v8f acc[4][4];
#pragma unroll
    for (int i = 0; i < 4; ++i)
#pragma unroll
        for (int j = 0; j < 4; ++j) acc[i][j] = (v8f){};

    // ---- prologue: stage chunk 0 ----
    {
        const float* ga = A + (size_t)srow * MD + scol;
        const float* gb = B + (size_t)srow * MD + scol;
        unsigned la = (unsigned)(uintptr_t)(&lds[stW]);
        unsigned lb = (unsigned)(uintptr_t)(&lds[CHW + stW]);
#pragma unroll
        for (int q = 0; q < 4; ++q) {
            async_ld16(la + 16 * q, ga + 4 * q);
            async_ld16(lb + 16 * q, gb + 4 * q);
        }
    }
    wait_async0();
    __syncthreads();

    // ---- main pipeline over 16 K-chunks ----
    for (int c = 0; c < NCHUNK; ++c) {
        const int buf = c & 1;

        if (c + 1 < NCHUNK) { // prefetch next chunk into other buffer
            const int k0 = (c + 1) * KC;
            const float* ga = A + (size_t)(k0 + srow) * MD + scol;
            const float* gb = B + (size_t)(k0 + srow) * MD + scol;
            unsigned la = (unsigned)(uintptr_t)(&lds[(buf ^ 1) * 2 * CHW + stW]);
            unsigned lb = (unsigned)(uintptr_t)(&lds[(buf ^ 1) * 2 * CHW + CHW + stW]);
#pragma unroll
            for (int q = 0; q < 4; ++q) {
                async_ld16(la + 16 * q, ga + 4 * q);
                async_ld16(lb + 16 * q, gb + 4 * q);
            }
        }

        // per-wave fragment base addresses for this buffer
        const unsigned rowOff = (unsigned)((2 * laneHi) * STRIDE + laneLo);
        unsigned addrA = (unsigned)(uintptr_t)(&lds[buf * 2 * CHW + rowOff + 64 * wm]);
        unsigned addrB = (unsigned)(uintptr_t)(&lds[buf * 2 * CHW + CHW + rowOff + 64 * wn]);

#pragma unroll
        for (int kk = 0; kk < KC; kk += 4) {
            v2f a0, a1, a2, a3, b0, b1, b2, b3;
            frag_ld(addrA, addrB, a0, a1, a2, a3, b0, b1, b2, b3);
            addrA += 4 * STRIDE * 4; // next 4 k-rows (bytes)
            addrB += 4 * STRIDE * 4;

            v2f af[4] = {a0, a1, a2, a3};
            v2f bf[4] = {b0, b1, b2, b3};
#pragma unroll
            for (int i = 0; i < 4; ++i)
#pragma unroll
                for (int j = 0; j < 4; ++j)
                    acc[i][j] = __builtin_amdgcn_wmma_f32_16x16x4_f32(
                        false, af[i], false, bf[j], (short)0, acc[i][j],
                        false, false);
        }

        wait_async0();     // next chunk landed (own wave's ops)
        __syncthreads();   // all waves done reading `buf` + all writes visible
    }

    // ---- row min/max over all 16384 scores (128 values per lane) ----
    float vmin = 3.402823466e+38f, vmax = -3.402823466e+38f;
#pragma unroll
    for (int i = 0; i < 4; ++i)
#pragma unroll
        for (int j = 0; j < 4; ++j)
#pragma unroll
            for (int r = 0; r < 8; ++r) {
                float v = acc[i][j][r];
                vmin = fminf(vmin, v);
                vmax = fmaxf(vmax, v);
            }
#pragma unroll
    for (int off = 16; off >= 1; off >>= 1) {
        vmin = fminf(vmin, __shfl_xor(vmin, off, 32));
        vmax = fmaxf(vmax, __shfl_xor(vmax, off, 32));
    }
    if (lane == 0) { sMin[w] = vmin; sMax[w] = vmax; }
    if (tid < NBIN) sBins[tid] = 0;
    __syncthreads();

    float mn = sMin[0], mx = sMax[0];
#pragma unroll
    for (int i = 1; i < 4; ++i) {
        mn = fminf(mn, sMin[i]);
        mx = fmaxf(mx, sMax[i]);
    }
    const float denom = (mx > mn) ? (mx - mn) : 1.0f;
    const float scale = (float)NBIN / denom;

    // ---- bin 128 values/lane into packed 8x8-bit counters (max 128/bin) ----
    unsigned long long pack = 0ull;
#pragma unroll
    for (int i = 0; i < 4; ++i)
#pragma unroll
        for (int j = 0; j < 4; ++j)
#pragma unroll
            for (int r = 0; r < 8; ++r) {
                float v  = acc[i][j][r];
                int  idx = (int)floorf((v - mn) * scale);
                idx = idx < 0 ? 0 : (idx > NBIN - 1 ? NBIN - 1 : idx);
                pack += 1ull << (idx * 8);
            }

#pragma unroll
    for (int b = 0; b < NBIN; ++b) {
        int cb = (int)((pack >> (8 * b)) & 0xFFull);
#pragma unroll
        for (int off = 16; off >= 1; off >>= 1) cb += __shfl_xor(cb, off, 32);
        if (lane == 0) atomicAdd(&sBins[b], cb);
    }
    __syncthreads();

    // ---- L2 normalize 8-bin histogram, write out[p, t*8 + bin] ----
    if (tid < NBIN) {
        float ss = 0.0f;
#pragma unroll
        for (int b = 0; b < NBIN; ++b) {
            float h = (float)sBins[b];
            ss += h * h;
        }
        float nrm = fmaxf(sqrtf(ss), 1e-12f);
        out[(size_t)p * 16 + t * NBIN + tid] = (float)sBins[tid] / nrm;
    }
}

extern "C" void kernel_launch(void* const* d_in, const int* in_sizes, int n_in,
                              void* d_out, int out_size, void* d_ws, size_t ws_size,
                              hipStream_t stream) {
    const float* m1  = (const float*)d_in[0];
    const float* m2  = (const float*)d_in[1];
    float*       out = (float*)d_out;
    (void)in_sizes; (void)n_in; (void)out_size; (void)d_ws; (void)ws_size;

    dim3 grid(2 * BS * BS); // 8192 workgroups: (input, a, b)
    dim3 block(128);        // 4 wave32 waves, 64x64 score quadrant each
    hipLaunchKernelGGL(gram_hist_kernel, grid, block, 0, stream, m1, m2, out);
}